// Attention_16801912062520
// MI455X (gfx1250) — compile-verified
//
#include <hip/hip_runtime.h>
#include <hip/hip_bf16.h>

// ---------------------------------------------------------------------------
// Problem: dps=32, seq=2048, d=1024, H=16, DK=64, f32 in/out.
// Algebra: beta[b,s,h] = <S4[b,s,h,:], w[b,h,:]> + c[b,h]
//          w[b,h,:]    = R4[b,h,:] @ M + mb,      M = WK_w^T @ WQ_w
//          mb[d]       = sum_e WQ_w[e,d]*WK_b[e]
//          c[b,h]      = R4[b,h,:] @ q + q0,      q[d'] = sum_e WQ_b[e]*WK_w[e,d']
//          rv[b,h,:]   = R4[b,h,:] @ WV_w^T + WV_b
//          out         = S_mas ? beta * rv : 0
// Main kernel is pure HBM streaming (~384MB -> ~16.5us @ 23.3TB/s).
// S / out are single-touch 256MB streams (> 192MB L2) -> non-temporal hints.
// Precompute GEMMs (64x64x64 and 32x(16x64x64)) use V_WMMA_F32_16X16X4_F32.
// ---------------------------------------------------------------------------

typedef __attribute__((ext_vector_type(2))) float v2f;
typedef __attribute__((ext_vector_type(4))) float v4f;   // native vector: OK for
                                                         // nontemporal builtins
typedef __attribute__((ext_vector_type(8))) float v8f;

// WMMA f32 16x16x4 lane layouts (ISA 7.12.2, wave32):
//   A (16x4): lane L, r=L&15, half=L>>4:  a.x=A[r][k0+2*half], a.y=A[r][k0+2*half+1]
//   B (4x16): col c=L&15:                 b.x=B[k0+2*half][c], b.y=B[k0+2*half+1][c]
//   D (16x16): vgpr v:                    D[v + 8*half][c]

__device__ __forceinline__ v8f wmma_f32_k4(v2f a, v2f b, v8f c) {
    return __builtin_amdgcn_wmma_f32_16x16x4_f32(false, a, false, b, (short)0, c,
                                                 false, false);
}

// ---------------- Kernel P1: M = WK_w^T @ WQ_w, plus bias folds -------------
__global__ __launch_bounds__(32) void precompute_M(
    const float* __restrict__ WQ_w, const float* __restrict__ WQ_b,
    const float* __restrict__ WK_w, const float* __restrict__ WK_b,
    float* __restrict__ Mout, float* __restrict__ mb,
    float* __restrict__ q, float* __restrict__ q0)
{
    const int lane = threadIdx.x;      // one wave32
    const int half = lane >> 4;
    const int r    = lane & 15;

    // M = A x B with A[d'][e] = WK_w[e][d'], B[e][d] = WQ_w[e][d]  (64x64, K=64)
    for (int ti = 0; ti < 4; ++ti) {
        for (int tj = 0; tj < 4; ++tj) {
            v8f acc = {};
            #pragma unroll
            for (int k = 0; k < 16; ++k) {
                const int e0 = k * 4 + 2 * half;
                v2f a, b;
                a.x = WK_w[(e0 + 0) * 64 + (ti * 16 + r)];
                a.y = WK_w[(e0 + 1) * 64 + (ti * 16 + r)];
                b.x = WQ_w[(e0 + 0) * 64 + (tj * 16 + r)];
                b.y = WQ_w[(e0 + 1) * 64 + (tj * 16 + r)];
                acc = wmma_f32_k4(a, b, acc);
            }
            #pragma unroll
            for (int v = 0; v < 8; ++v) {
                const int m = ti * 16 + v + 8 * half;
                const int n = tj * 16 + r;
                Mout[m * 64 + n] = acc[v];
            }
        }
    }

    // scalar bias folds (all zero in practice, computed for generality)
    for (int t = 0; t < 2; ++t) {
        const int d = lane * 2 + t;
        float smb = 0.f, sq = 0.f;
        for (int e = 0; e < 64; ++e) {
            smb += WQ_w[e * 64 + d] * WK_b[e];
            sq  += WQ_b[e] * WK_w[e * 64 + d];
        }
        mb[d] = smb;
        q[d]  = sq;
    }
    if (lane == 0) {
        float s = 0.f;
        for (int e = 0; e < 64; ++e) s += WQ_b[e] * WK_b[e];
        q0[0] = s;
    }
}

// ------------- Kernel P2: per-batch w, rv, c (one wave per batch) -----------
__global__ __launch_bounds__(32) void precompute_batch(
    const float* __restrict__ R, const float* __restrict__ Mmat,
    const float* __restrict__ mb, const float* __restrict__ WV_w,
    const float* __restrict__ WV_b, const float* __restrict__ q,
    const float* __restrict__ q0,
    float* __restrict__ w_all, float* __restrict__ rv_all,
    float* __restrict__ c_all)
{
    const int b    = blockIdx.x;
    const int lane = threadIdx.x;
    const int half = lane >> 4;
    const int r    = lane & 15;
    const float* Rb = R + (size_t)b * 1024;   // A[h][d], 16x64 row-major

    // w = A @ M + mb ; rv = A @ WV_w^T + WV_b    (M-dim = 16 heads -> 1 tile row)
    for (int tj = 0; tj < 4; ++tj) {
        v8f accW = {}, accV = {};
        #pragma unroll
        for (int k = 0; k < 16; ++k) {
            const int k0 = k * 4 + 2 * half;
            v2f a, bw, bv;
            a.x  = Rb[r * 64 + k0];
            a.y  = Rb[r * 64 + k0 + 1];
            bw.x = Mmat[(k0 + 0) * 64 + tj * 16 + r];          // B = M
            bw.y = Mmat[(k0 + 1) * 64 + tj * 16 + r];
            bv.x = WV_w[(tj * 16 + r) * 64 + k0];              // B = WV_w^T
            bv.y = WV_w[(tj * 16 + r) * 64 + k0 + 1];
            accW = wmma_f32_k4(a, bw, accW);
            accV = wmma_f32_k4(a, bv, accV);
        }
        #pragma unroll
        for (int v = 0; v < 8; ++v) {
            const int h = v + 8 * half;       // head (D-matrix row)
            const int n = tj * 16 + r;        // dk   (D-matrix col)
            w_all [(b * 16 + h) * 64 + n] = accW[v] + mb[n];
            rv_all[(b * 16 + h) * 64 + n] = accV[v] + WV_b[n];
        }
    }

    // c[b,h] after WMMA region (keep EXEC all-1s during WMMAs)
    if (lane < 16) {
        float s = q0[0];
        for (int d = 0; d < 64; ++d) s += Rb[lane * 64 + d] * q[d];
        c_all[b * 16 + lane] = s;
    }
}

// ----------------- Main streaming kernel: HBM-bound gate --------------------
// grid (32 batches, 32), block 256 (8 waves). Wave handles one token per
// iteration: 8 x v4f per lane covers the token's 1024 floats coalesced.
// v4f index i*32+lane -> element i*128+lane*4 -> head 2*i+(lane>>4),
// dk = (lane&15)*4. Reduce dot over each 16-lane head group via shfl_xor.
// S loads and out stores carry non-temporal hints (single-touch, > L2 size).
__global__ __launch_bounds__(256) void fused_gate(
    const float* __restrict__ S, const int* __restrict__ S_mas,
    const float* __restrict__ w_all, const float* __restrict__ rv_all,
    const float* __restrict__ c_all, float* __restrict__ out)
{
    __shared__ float s_w [16 * 64];
    __shared__ float s_rv[16 * 64];
    __shared__ float s_c [16];

    const int b   = blockIdx.x;
    const int tid = threadIdx.x;
    for (int i = tid; i < 1024; i += 256) {
        s_w [i] = w_all [b * 1024 + i];
        s_rv[i] = rv_all[b * 1024 + i];
    }
    if (tid < 16) s_c[tid] = c_all[b * 16 + tid];
    __syncthreads();

    const int lane = tid & 31;
    const int wid  = tid >> 5;
    const int half = lane >> 4;
    const int r    = lane & 15;

    for (int s = blockIdx.y * 8 + wid; s < 2048; s += gridDim.y * 8) {
        const size_t base = ((size_t)(b * 2048 + s)) * 1024;
        const v4f* Sp = (const v4f*)(S + base);
        v4f*       Op = (v4f*)(out + base);
        const bool mask = (S_mas[b * 2048 + s] != 0);

        if (!mask) {
            // masked token: skip S reads entirely, just store zeros
            const v4f z = {0.f, 0.f, 0.f, 0.f};
            #pragma unroll
            for (int i = 0; i < 8; ++i)
                __builtin_nontemporal_store(z, &Op[i * 32 + lane]);
            continue;
        }
        #pragma unroll
        for (int i = 0; i < 8; ++i) {
            const int h = 2 * i + half;
            const v4f sv = __builtin_nontemporal_load(&Sp[i * 32 + lane]);
            const v4f wv = *(const v4f*)(&s_w[h * 64 + r * 4]);
            float p = sv.x * wv.x + sv.y * wv.y + sv.z * wv.z + sv.w * wv.w;
            p += __shfl_xor(p, 1);
            p += __shfl_xor(p, 2);
            p += __shfl_xor(p, 4);
            p += __shfl_xor(p, 8);
            const float beta = p + s_c[h];
            const v4f rv = *(const v4f*)(&s_rv[h * 64 + r * 4]);
            v4f o;
            o.x = beta * rv.x; o.y = beta * rv.y;
            o.z = beta * rv.z; o.w = beta * rv.w;
            __builtin_nontemporal_store(o, &Op[i * 32 + lane]);
        }
    }
}

// ---------------------------------------------------------------------------
extern "C" void kernel_launch(void* const* d_in, const int* in_sizes, int n_in,
                              void* d_out, int out_size, void* d_ws, size_t ws_size,
                              hipStream_t stream) {
    const float* S     = (const float*)d_in[0];
    const float* R     = (const float*)d_in[1];
    const int*   S_mas = (const int*)  d_in[2];
    // d_in[3] (R_mas) is dead in the reference
    const float* WQ_w  = (const float*)d_in[4];
    const float* WQ_b  = (const float*)d_in[5];
    const float* WK_w  = (const float*)d_in[6];
    const float* WK_b  = (const float*)d_in[7];
    const float* WV_w  = (const float*)d_in[8];
    const float* WV_b  = (const float*)d_in[9];
    float* out = (float*)d_out;

    // workspace layout (floats): M[4096] mb[64] q[64] q0[1..pad] w[32768]
    // rv[32768] c[512]  -> ~282 KB total
    float* ws     = (float*)d_ws;
    float* Mmat   = ws;
    float* mb     = ws + 4096;
    float* q      = ws + 4160;
    float* q0     = ws + 4224;
    float* w_all  = ws + 4352;
    float* rv_all = ws + 4352 + 32768;
    float* c_all  = ws + 4352 + 65536;

    precompute_M<<<1, 32, 0, stream>>>(WQ_w, WQ_b, WK_w, WK_b, Mmat, mb, q, q0);
    precompute_batch<<<32, 32, 0, stream>>>(R, Mmat, mb, WV_w, WV_b, q, q0,
                                            w_all, rv_all, c_all);
    dim3 grid(32, 32);
    fused_gate<<<grid, 256, 0, stream>>>(S, S_mas, w_all, rv_all, c_all, out);
}